// Control_Loss_31550829756871
// MI455X (gfx1250) — compile-verified
//
#include <hip/hip_runtime.h>
#include <hip/hip_bf16.h>
#include <cstdint>
#include <cstddef>

// out = support[0] + 0.01 * sum_rows sum( |m| where |m| > sorted(|m|)[2048] )
// Rows: B=8192, L=4096. One block (256 thr = 8 wave32) per row.
// Row staged in LDS via gfx1250 async global->LDS loads (ASYNCcnt path),
// 4x8-bit radix select on |m| bit patterns with wave-shuffle scans
// (minimal barriers), deterministic reductions (no float atomics).

#define ROW_L      4096
#define BLOCK_T    256
#define N_WAVES    (BLOCK_T / 32)      // 8
#define PER_THREAD (ROW_L / BLOCK_T)   // 16 elements / thread
#define CHUNKS     (ROW_L * 4 / 16)    // 1024 16-byte chunks per row
#define THR_IDX    2048                // int(L * (1 - 0.5))

typedef int v4i_cl __attribute__((vector_size(16)));

// ---- gfx1250 async global->LDS copy (16 bytes per lane) -------------------
__device__ __forceinline__ void async_copy16(const void* gsrc, void* lds_dst) {
#if __has_builtin(__builtin_amdgcn_global_load_async_to_lds_b128)
    // (global v4i*, lds v4i*, imm offset, imm cpol) -- codegen-verified
    __builtin_amdgcn_global_load_async_to_lds_b128(
        (v4i_cl*)gsrc, (v4i_cl*)lds_dst, 0, 0);
#else
    asm volatile("global_load_async_to_lds_b128 %0, %1, off"
                 :
                 : "v"((unsigned)(uintptr_t)lds_dst),
                   "v"((unsigned long long)(uintptr_t)gsrc)
                 : "memory");
#endif
}

__device__ __forceinline__ void wait_async0(void) {
#if __has_builtin(__builtin_amdgcn_s_wait_asynccnt)
    __builtin_amdgcn_s_wait_asynccnt(0);
#else
    asm volatile("s_wait_asynccnt 0" ::: "memory");
#endif
}

// ---------------------------------------------------------------------------
__global__ __launch_bounds__(BLOCK_T) void row_select_sum_kernel(
        const float* __restrict__ masks, float* __restrict__ row_sums) {
    __shared__ __align__(16) unsigned sdata[ROW_L];   // raw float bits of row
    __shared__ unsigned hist[256];
    __shared__ unsigned wsum[N_WAVES];                // per-wave scan totals
    __shared__ unsigned bcast[2];                     // chosen bin, new rank
    __shared__ float    red[N_WAVES];

    const int      t     = threadIdx.x;
    const int      lane  = t & 31;
    const int      w     = t >> 5;
    const unsigned row   = blockIdx.x;
    const char*    growp = (const char*)(masks + (size_t)row * ROW_L);

    // ---- stage row into LDS: 1024 chunks of 16B, 4 per thread -------------
    #pragma unroll
    for (int i = 0; i < CHUNKS / BLOCK_T; ++i) {
        const int c = i * BLOCK_T + t;                // contiguous, coalesced
        async_copy16(growp + (size_t)c * 16, (void*)&sdata[c * 4]);
    }
    wait_async0();         // this wave's async copies have landed in LDS
    __syncthreads();       // all waves' copies visible to everyone

    // ---- radix select: k-th smallest |m| bit pattern (k = THR_IDX) --------
    unsigned prefix = 0u;
    unsigned k      = THR_IDX;
    #pragma unroll
    for (int shift = 24; shift >= 0; shift -= 8) {
        hist[t] = 0u;
        __syncthreads();

        const unsigned settled = (shift == 24) ? 0u : (0xFFFFFFFFu << (shift + 8));
        #pragma unroll
        for (int j = 0; j < PER_THREAD; ++j) {
            const unsigned key = sdata[j * BLOCK_T + t] & 0x7FFFFFFFu;  // |m| bits
            if ((key & settled) == prefix)
                atomicAdd(&hist[(key >> shift) & 0xFFu], 1u);
        }
        __syncthreads();

        // 256-bin scan: wave-local inclusive scan (no barriers) + wave offsets
        const unsigned cnt = hist[t];
        unsigned incl = cnt;
        #pragma unroll
        for (int d = 1; d < 32; d <<= 1) {
            const unsigned u = __shfl_up(incl, d, 32);
            if (lane >= d) incl += u;
        }
        if (lane == 31) wsum[w] = incl;   // wave total
        __syncthreads();
        unsigned offset = 0u;
        #pragma unroll
        for (int ww = 0; ww < N_WAVES; ++ww)
            if (ww < w) offset += wsum[ww];
        const unsigned gincl = incl + offset;
        const unsigned gexcl = gincl - cnt;
        if (k >= gexcl && k < gincl) {    // exactly one bin matches
            bcast[0] = (unsigned)t;
            bcast[1] = k - gexcl;
        }
        __syncthreads();
        prefix |= bcast[0] << shift;
        k       = bcast[1];
        __syncthreads();
    }
    const unsigned thr_bits = prefix;     // exact bits of sorted(|m|)[2048]

    // ---- sum of |m| strictly above threshold (fixed order => deterministic)
    float s = 0.0f;
    #pragma unroll
    for (int j = 0; j < PER_THREAD; ++j) {
        const unsigned key = sdata[j * BLOCK_T + t] & 0x7FFFFFFFu;
        if (key > thr_bits) s += __uint_as_float(key);
    }
    #pragma unroll
    for (int d = 16; d >= 1; d >>= 1) s += __shfl_down(s, d, 32);  // wave reduce
    if (lane == 0) red[w] = s;
    __syncthreads();
    if (t == 0) {
        float tot = 0.0f;
        #pragma unroll
        for (int ww = 0; ww < N_WAVES; ++ww) tot += red[ww];       // fixed order
        row_sums[row] = tot;
    }
}

// ---------------------------------------------------------------------------
__global__ __launch_bounds__(BLOCK_T) void final_reduce_kernel(
        const float* __restrict__ row_sums, const float* __restrict__ support,
        float* __restrict__ out, int n_rows) {
    __shared__ float red[N_WAVES];
    const int t    = threadIdx.x;
    const int lane = t & 31;
    const int w    = t >> 5;
    float s = 0.0f;
    for (int i = t; i < n_rows; i += BLOCK_T) s += row_sums[i];    // fixed order
    #pragma unroll
    for (int d = 16; d >= 1; d >>= 1) s += __shfl_down(s, d, 32);
    if (lane == 0) red[w] = s;
    __syncthreads();
    if (t == 0) {
        float tot = 0.0f;
        #pragma unroll
        for (int ww = 0; ww < N_WAVES; ++ww) tot += red[ww];
        out[0] = support[0] + 0.01f * tot;
    }
}

// ---------------------------------------------------------------------------
extern "C" void kernel_launch(void* const* d_in, const int* in_sizes, int n_in,
                              void* d_out, int out_size, void* d_ws, size_t ws_size,
                              hipStream_t stream) {
    const float* support = (const float*)d_in[0];   // outputs_support, 1 elem
    // d_in[1] outputs_delete, d_in[2] targets: unused by the reference result
    const float* masks   = (const float*)d_in[3];   // [B, 4096] f32
    const int    n_rows  = in_sizes[3] / ROW_L;     // 8192

    float* row_sums = (float*)d_ws;                 // n_rows floats of scratch

    row_select_sum_kernel<<<n_rows, BLOCK_T, 0, stream>>>(masks, row_sums);
    final_reduce_kernel<<<1, BLOCK_T, 0, stream>>>(row_sums, support,
                                                   (float*)d_out, n_rows);
}